// FlashDepthWiseConv1dPython_1846835937819
// MI455X (gfx1250) — compile-verified
//
#include <hip/hip_runtime.h>
#include <hip/hip_bf16.h>
#include <stdint.h>

// Depthwise conv1d (B=4, D=2048, L=8192, K=3, pad=1), fp32.
// HBM-bound (0.75 FLOP/byte): TDM async global->LDS tile copies (free OOB
// zero-fill for the right halo), LDS-sourced FMAs, non-temporal b128 output
// stores (streamed 256 MiB output > 192 MB L2). All dims compile-time so the
// pre-TDM scalar preamble is shifts/masks, not a runtime divide.

typedef float v4f  __attribute__((ext_vector_type(4)));
typedef unsigned int v4u __attribute__((ext_vector_type(4)));
typedef int   v8i  __attribute__((ext_vector_type(8)));
typedef int   v4i  __attribute__((ext_vector_type(4)));

constexpr int kB          = 4;
constexpr int kD          = 2048;   // power of two -> mask
constexpr int kL          = 8192;
constexpr int kTile       = 2048;
constexpr int kThreads    = 256;
constexpr int kPerThread  = kTile / kThreads;            // 8
constexpr int kTilesPerRow = kL / kTile;                 // 4 (power of two -> shift)
constexpr int kTileShift  = 2;                           // log2(kTilesPerRow)

static_assert((kD & (kD - 1)) == 0 && (kTilesPerRow & (kTilesPerRow - 1)) == 0, "pow2");

__global__ __launch_bounds__(kThreads)
void dwconv1d_k3_tdm(const float* __restrict__ in,
                     const float* __restrict__ w,
                     const float* __restrict__ bias,
                     float* __restrict__ out)
{
    __shared__ float smem[kTile + 2];   // tile + 2-element halo

    const int block = blockIdx.x;
    const int tile  = block & (kTilesPerRow - 1);
    const int row   = block >> kTileShift;     // row = b*D + d
    const int d     = row & (kD - 1);

    const int tile_start = tile * kTile;
    int src_start = tile_start - 1;            // first input element needed (may be -1)
    int lshift    = 0;
    if (src_start < 0) { src_start = 0; lshift = 1; }   // left edge: shift copy, zero-pad the first LDS slot
    const unsigned n_copy      = (unsigned)(kTile + 2 - lshift); // elements TDM moves
    const unsigned tensor_dim0 = (unsigned)(kL - src_start);     // OOB -> zero (right halo)

    const uint64_t gaddr = (uint64_t)(uintptr_t)(in + (size_t)row * (size_t)kL + (size_t)src_start);
    // Generic LDS pointer: low 32 bits are the in-allocation byte offset.
    const unsigned lds_byte = (unsigned)(uintptr_t)(&smem[0]) + (unsigned)lshift * 4u;

    if (threadIdx.x < 32u) {   // wave 0 issues the TDM descriptor (EXEC ignored by TDM)
        // ---- D# group 0 (128b): count=1, lds_addr, global_addr[56:0], type=2 ----
        v4u g0;
        g0[0] = 1u;                                          // count=1, user descriptor
        g0[1] = lds_byte;                                    // lds_addr (bytes)
        g0[2] = (unsigned)(gaddr & 0xFFFFFFFFull);           // global_addr[31:0]
        g0[3] = (unsigned)((gaddr >> 32) & 0x01FFFFFFull)    // global_addr[56:32]
              | (2u << 30);                                  // type = 2 ("image")
        // ---- D# group 1 (256b): data_size=4B, dims/strides, 1-D tile ----
        v8i g1;
        g1[0] = (int)(2u << 16);                             // data_size = 4 bytes
        g1[1] = (int)((tensor_dim0 & 0xFFFFu) << 16);        // tensor_dim0[15:0]
        g1[2] = (int)(((tensor_dim0 >> 16) & 0xFFFFu)        // tensor_dim0[31:16]
              | (1u << 16));                                 // tensor_dim1 = 1
        g1[3] = (int)(n_copy << 16);                         // tile_dim0
        g1[4] = 1;                                           // tile_dim1 = 1
        g1[5] = kL;                                          // tensor_dim0_stride
        g1[6] = 0;
        g1[7] = 0;
        v4i gz  = {0, 0, 0, 0};                              // groups 2/3 unused (<=2D)
        v8i gz8 = {0, 0, 0, 0, 0, 0, 0, 0};                  // trailing group operand (unused)
        __builtin_amdgcn_tensor_load_to_lds(g0, g1, gz, gz, gz8, /*cpol=*/0);
        if (lshift && threadIdx.x == 0) smem[0] = 0.0f;      // zero-pad for l = -1
        __builtin_amdgcn_s_wait_tensorcnt(0);
    }
    __syncthreads();

    const float w0 = w[d * 3 + 0];
    const float w1 = w[d * 3 + 1];
    const float w2 = w[d * 3 + 2];
    const float bb = bias[d];

    // smem[j] holds in[row, tile_start - 1 + j]; out[l] = b + w0*x[l-1] + w1*x[l] + w2*x[l+1]
    const int t = (int)threadIdx.x * kPerThread;
    float s[kPerThread + 2];
#pragma unroll
    for (int i = 0; i < kPerThread + 2; ++i) s[i] = smem[t + i];

    v4f o0, o1;
#pragma unroll
    for (int i = 0; i < 4; ++i)
        o0[i] = __builtin_fmaf(w2, s[i + 2],
                __builtin_fmaf(w1, s[i + 1],
                __builtin_fmaf(w0, s[i + 0], bb)));
#pragma unroll
    for (int i = 0; i < 4; ++i)
        o1[i] = __builtin_fmaf(w2, s[i + 6],
                __builtin_fmaf(w1, s[i + 5],
                __builtin_fmaf(w0, s[i + 4], bb)));

    float* op = out + (size_t)row * (size_t)kL + (size_t)(tile_start + t);
    __builtin_nontemporal_store(o0, (v4f*)op);        // streamed output: b128 NT stores
    __builtin_nontemporal_store(o1, (v4f*)(op + 4));
}

extern "C" void kernel_launch(void* const* d_in, const int* in_sizes, int n_in,
                              void* d_out, int out_size, void* d_ws, size_t ws_size,
                              hipStream_t stream) {
    const float* in = (const float*)d_in[0];   // [B, D, L] fp32
    const float* w  = (const float*)d_in[1];   // [D, 3]    fp32
    const float* b  = (const float*)d_in[2];   // [D]       fp32
    // d_in[3] is padding (=1) on device; K=3/pad=1 are baked into the kernel.
    float* out = (float*)d_out;                // [B, D, L] fp32 (L_out == L)

    dim3 grid((unsigned)(kB * kD * kTilesPerRow));   // 32768 blocks
    dwconv1d_k3_tdm<<<grid, kThreads, 0, stream>>>(in, w, b, out);
}